// Optimized_Phi_Layer_v3_88244398063838
// MI455X (gfx1250) — compile-verified
//
#include <hip/hip_runtime.h>
#include <hip/hip_bf16.h>
#include <math.h>

// Problem constants (B=8, Mr=32, Nr=64, G=64, NUM_SC=32)
// Output layout (floats, concatenated):
//   y_ri   : [8,1024,2]       off 0          len 16384
//   Phi_ri : [8,1024,4096,2]  off 16384      len 67108864
//   A_R_ri : [8,64,64,2]      off 67125248   len 65536
//   A_K_ri : [8,32,64,2]      off 67190784   len 32768
static constexpr size_t Y_OFF   = 0;
static constexpr size_t PHI_OFF = 16384;
static constexpr size_t AR_OFF  = 67125248;
static constexpr size_t AK_OFF  = 67190784;
static constexpr size_t PHI_BSTRIDE = 8388608;  // 1024*4096*2 floats per batch

// Workspace layout (floats):
//   BT_r[64*32] @ 0      : WH^T real  = cos(W[n,m])/8     (index n*32+m)
//   BT_i[64*32] @ 2048   : WH^T imag  = -sin(W[n,m])/8
//   ARr [64*64] @ 4096, ARi @ 8192
//   AKr [32*64] @ 12288, AKi @ 14336
//   WAr [32*64] @ 16384, WAi @ 18432
static constexpr int WS_BTR = 0,   WS_BTI = 2048;
static constexpr int WS_ARR = 4096, WS_ARI = 8192;
static constexpr int WS_AKR = 12288, WS_AKI = 14336;
static constexpr int WS_WAR = 16384, WS_WAI = 18432;

typedef __attribute__((ext_vector_type(2))) float v2f;
typedef __attribute__((ext_vector_type(4))) float v4f;
typedef __attribute__((ext_vector_type(8))) float v8f;

// ---------------------------------------------------------------------------
// Kernel 1: build trig dictionaries, WH^T; emit A_R_ri / A_K_ri outputs (x8 batch)
// 4096 threads total.
// ---------------------------------------------------------------------------
__global__ __launch_bounds__(256) void k_prep(const float* __restrict__ W,
                                              const float* __restrict__ AR,
                                              const float* __restrict__ AK,
                                              float* __restrict__ ws,
                                              float* __restrict__ out) {
  const float PI = 3.14159265358979323846f;
  int t = blockIdx.x * blockDim.x + threadIdx.x;
  if (t < 4096) {                       // A_R: n = t>>6, g = t&63
    int n = t >> 6, g = t & 63;
    float ang = PI * (float)n * AR[g];
    float cr = cosf(ang) * 0.125f;      // 1/sqrt(64)
    float ci = -sinf(ang) * 0.125f;
    ws[WS_ARR + t] = cr;
    ws[WS_ARI + t] = ci;
    v2f v; v.x = cr; v.y = ci;
#pragma unroll
    for (int b = 0; b < 8; ++b)
      *(v2f*)(out + AR_OFF + (size_t)b * 8192 + (size_t)t * 2) = v;
  }
  if (t < 2048) {                       // A_K: k = t>>6, g = t&63
    int k = t >> 6, g = t & 63;
    float ang = PI * (float)k * AK[g];
    const float inv = 0.17677669529663688f;  // 1/sqrt(32)
    float cr = cosf(ang) * inv;
    float ci = -sinf(ang) * inv;
    ws[WS_AKR + t] = cr;
    ws[WS_AKI + t] = ci;
    v2f v; v.x = cr; v.y = ci;
#pragma unroll
    for (int b = 0; b < 8; ++b)
      *(v2f*)(out + AK_OFF + (size_t)b * 4096 + (size_t)t * 2) = v;
    // WH^T (same flat layout as W: [n,m] = t)
    float w = W[t];
    ws[WS_BTR + t] = cosf(w) * 0.125f;
    ws[WS_BTI + t] = -sinf(w) * 0.125f;
  }
}

// ---------------------------------------------------------------------------
// Kernel 2: WA[m,h] = sum_n WH[m,n] * A_R[n,h]   (2048 threads, 64-deep)
// ---------------------------------------------------------------------------
__global__ __launch_bounds__(256) void k_wa(float* __restrict__ ws) {
  int t = blockIdx.x * blockDim.x + threadIdx.x;
  if (t >= 2048) return;
  int m = t >> 6, h = t & 63;
  float sr = 0.f, si = 0.f;
#pragma unroll 4
  for (int n = 0; n < 64; ++n) {
    float whr = ws[WS_BTR + n * 32 + m];
    float whi = ws[WS_BTI + n * 32 + m];
    float arr = ws[WS_ARR + n * 64 + h];
    float ari = ws[WS_ARI + n * 64 + h];
    sr += whr * arr - whi * ari;
    si += whr * ari + whi * arr;
  }
  ws[WS_WAR + t] = sr;
  ws[WS_WAI + t] = si;
}

// ---------------------------------------------------------------------------
// Kernel 3: y[b,s,m] = sum_n WH[m,n]*(h+noise)[b,s,n] via V_WMMA_F32_16X16X4_F32.
// Rows r = b*32+s (256), cols m (32). One wave (32 threads) per 16x16 tile:
// grid = 16 row-tiles * 2 col-tiles = 32 blocks. Complex GEMM = 4 real WMMA
// chains. EXEC all-1s (blockDim==32, no divergence) as WMMA requires.
// A 16x4 f32 layout: lane L -> M=L%16, VGPR j -> K = 2*(L/16)+j.
// B 4x16 layout:     lane L -> N=L%16, VGPR j -> K = 2*(L/16)+j.
// D 16x16 layout:    lane L, VGPR v -> M = v+8*(L/16), N = L%16.
// ---------------------------------------------------------------------------
__global__ __launch_bounds__(32) void k_y(const float* __restrict__ h,
                                          const float* __restrict__ noise,
                                          const float* __restrict__ ws,
                                          float* __restrict__ out) {
  int lane = threadIdx.x;
  int rt = blockIdx.x >> 1;         // row tile 0..15
  int ct = blockIdx.x & 1;          // col tile 0..1
  int half = lane >> 4;             // 0/1
  int l16 = lane & 15;

  int arow = rt * 16 + l16;         // global row = b*32 + s
  int ab = arow >> 5, as = arow & 31;
  size_t xbase = (size_t)ab * 4096 + (size_t)as * 128;  // float idx of (b, s*64, 0)
  int mcol = ct * 16 + l16;

  v8f acc_rr = {}, acc_ii = {}, acc_ri = {}, acc_ir = {};

  for (int k0 = 0; k0 < 16; ++k0) {
    v2f ar, ai, br, bi;
#pragma unroll
    for (int j = 0; j < 2; ++j) {
      int K = 4 * k0 + 2 * half + j;      // n index
      size_t xi = xbase + (size_t)K * 2;  // interleaved re/im
      v2f hv = *(const v2f*)(h + xi);
      v2f nv = *(const v2f*)(noise + xi);
      ar[j] = hv.x + nv.x;
      ai[j] = hv.y + nv.y;
      int bidx = K * 32 + mcol;           // WH^T[n, m]
      br[j] = ws[WS_BTR + bidx];
      bi[j] = ws[WS_BTI + bidx];
    }
    acc_rr = __builtin_amdgcn_wmma_f32_16x16x4_f32(false, ar, false, br, (short)0, acc_rr, false, false);
    acc_ii = __builtin_amdgcn_wmma_f32_16x16x4_f32(false, ai, false, bi, (short)0, acc_ii, false, false);
    acc_ri = __builtin_amdgcn_wmma_f32_16x16x4_f32(false, ar, false, bi, (short)0, acc_ri, false, false);
    acc_ir = __builtin_amdgcn_wmma_f32_16x16x4_f32(false, ai, false, br, (short)0, acc_ir, false, false);
  }

#pragma unroll
  for (int v = 0; v < 8; ++v) {
    int row = rt * 16 + v + 8 * half;
    int b = row >> 5, s = row & 31;
    v2f yv;
    yv.x = acc_rr[v] - acc_ii[v];   // real
    yv.y = acc_ri[v] + acc_ir[v];   // imag
    *(v2f*)(out + Y_OFF + (size_t)b * 2048 + ((size_t)s * 32 + mcol) * 2) = yv;
  }
}

// ---------------------------------------------------------------------------
// Kernel 4 (store-bound, ~269 MB): Phi[k,m,g,h] = A_K[k,g]*WA[m,h], written
// 8x (batch broadcast) with non-temporal B128 stores. One block per km row
// (1024 blocks x 256 threads); A_K row + WA row staged in LDS.
// ---------------------------------------------------------------------------
__global__ __launch_bounds__(256) void k_phi(const float* __restrict__ ws,
                                             float* __restrict__ out) {
  int km = blockIdx.x;              // 0..1023
  int k = km >> 5, m = km & 31;
  __shared__ float sAKr[64], sAKi[64], sWAr[64], sWAi[64];
  int t = threadIdx.x;
  if (t < 64) {
    sAKr[t] = ws[WS_AKR + k * 64 + t];
    sAKi[t] = ws[WS_AKI + k * 64 + t];
    sWAr[t] = ws[WS_WAR + m * 64 + t];
    sWAi[t] = ws[WS_WAI + m * 64 + t];
  }
  __syncthreads();

  size_t rowbase = PHI_OFF + (size_t)km * 8192;  // 4096 cplx = 8192 floats/row
#pragma unroll
  for (int i = 0; i < 8; ++i) {
    int p = i * 256 + t;            // pair index 0..2047 (two gh per thread)
    int gh0 = p * 2;
    int g = gh0 >> 6;               // gh0 even => gh0,gh0+1 share g
    int h0 = gh0 & 63, h1 = h0 + 1;
    float akr = sAKr[g], aki = sAKi[g];
    float war0 = sWAr[h0], wai0 = sWAi[h0];
    float war1 = sWAr[h1], wai1 = sWAi[h1];
    v4f v;
    v.x = akr * war0 - aki * wai0;  // re(gh0)
    v.y = akr * wai0 + aki * war0;  // im(gh0)
    v.z = akr * war1 - aki * wai1;  // re(gh1)
    v.w = akr * wai1 + aki * war1;  // im(gh1)
#pragma unroll
    for (int b = 0; b < 8; ++b) {
      __builtin_nontemporal_store(
          v, (v4f*)(out + rowbase + (size_t)b * PHI_BSTRIDE + (size_t)gh0 * 2));
    }
  }
}

// ---------------------------------------------------------------------------
extern "C" void kernel_launch(void* const* d_in, const int* in_sizes, int n_in,
                              void* d_out, int out_size, void* d_ws, size_t ws_size,
                              hipStream_t stream) {
  const float* h     = (const float*)d_in[0];
  const float* noise = (const float*)d_in[1];
  const float* W     = (const float*)d_in[2];
  const float* AR    = (const float*)d_in[3];
  const float* AK    = (const float*)d_in[4];
  float* out = (float*)d_out;
  float* ws  = (float*)d_ws;

  k_prep<<<16, 256, 0, stream>>>(W, AR, AK, ws, out);   // 4096 threads
  k_wa  <<<8, 256, 0, stream>>>(ws);                    // 2048 threads
  k_y   <<<32, 32, 0, stream>>>(h, noise, ws, out);     // 32 waves, WMMA
  k_phi <<<1024, 256, 0, stream>>>(ws, out);            // 269 MB NT stores
}